// CAN_29566554866256
// MI455X (gfx1250) — compile-verified
//
#include <hip/hip_runtime.h>
#include <math.h>

typedef __attribute__((ext_vector_type(2))) float v2f;
typedef __attribute__((ext_vector_type(8))) float v8f;

// Dims: B=16, IC=16, II=8, IDC=23, NC=32, NP=8, NI=8, P=1024, DC=23

// ---------------------------------------------------------------------------
// K0: build hat_a (WMMA f32 16x16x4) and hat_g (VALU) into workspace.
// grid = 8 (b-pairs) * 32 (nc) = 256 blocks, 128 threads (4 waves).
// hatA layout: [b][nc][p][16], hatG layout: [b][nc][p][8] (cols 0..5 valid)
// ---------------------------------------------------------------------------
__global__ __launch_bounds__(128) void can_hat_kernel(
    const float* __restrict__ x, const float* __restrict__ W1,
    const float* __restrict__ W2, float* __restrict__ hatA,
    float* __restrict__ hatG)
{
    __shared__ float Aat[16][16][16]; // [ic][m=(b_local*8+ii)][a]  attr
    __shared__ float Gp [16][16][8];  // [ic][m][e] gpose (e=6 -> 1.0)

    const int tid = threadIdx.x;
    const int b2  = blockIdx.x >> 5;   // batch pair index 0..7
    const int nc  = blockIdx.x & 31;

    for (int l = tid; l < 2 * 16 * 8 * 23; l += 128) {
        int dc = l % 23; int rest = l / 23;
        int ii = rest & 7; rest >>= 3;
        int ic = rest & 15; int bl = rest >> 4;
        int bg = b2 * 2 + bl;
        float v = x[((bg * 16 + ic) * 8 + ii) * 23 + dc];
        int m = bl * 8 + ii;
        if (dc >= 7)      Aat[ic][m][dc - 7] = v;   // attr   a = dc-7
        else if (dc >= 1) Gp [ic][m][dc - 1] = v;   // gpose  e = dc-1
    }
    for (int l = tid; l < 256; l += 128) {
        int ic = l >> 4, m = l & 15;
        Gp[ic][m][6] = 1.0f;
        Gp[ic][m][7] = 0.0f;
    }
    __syncthreads();

    const int lane = tid & 31;
    const int wv   = tid >> 5;
    const int rowc = lane & 15;   // A-row M (lanes<16) / B,C column N
    const int hi   = lane >> 4;   // lane half select

    // -------- apart: hat_a = attr * W2 via WMMA f32 16x16x4 ----------------
    for (int tt = wv; tt < 128; tt += 4) {           // (ic, np) tiles
        int ic = tt >> 3, np = tt & 7;
        const float* W2t = W2 + (((ic * 32 + nc) * 8 + np) << 8); // 16x16
        v8f c = {0.f, 0.f, 0.f, 0.f, 0.f, 0.f, 0.f, 0.f};
#pragma unroll
        for (int kk = 0; kk < 4; ++kk) {
            int k0 = kk * 4 + (hi << 1);
            v2f av = *(const v2f*)(&Aat[ic][rowc][k0]);
            v2f bv;
            bv.x = W2t[k0 * 16 + rowc];
            bv.y = W2t[(k0 + 1) * 16 + rowc];
            c = __builtin_amdgcn_wmma_f32_16x16x4_f32(
                    false, av, false, bv, (short)0, c, false, false);
        }
        int bg = b2 * 2 + hi;
        float* dst = hatA + (((bg * 32 + nc) * 1024 + np * 128 + ic * 8) << 4) + rowc;
#pragma unroll
        for (int r = 0; r < 8; ++r) dst[r << 4] = c[r];
    }

    // -------- gpart: hat_g = gpose * W1 (K=7, N=6) via VALU ----------------
    for (int rr = tid; rr < 2048; rr += 128) {
        int bl = rr >> 10; int p = rr & 1023;
        int np = p >> 7, ic = (p >> 3) & 15, ii = p & 7;
        int m  = bl * 8 + ii;
        int bg = b2 * 2 + bl;
        const float* W1t = W1 + ((ic * 32 + nc) * 8 + np) * 42; // 7x6
        float g0 = Gp[ic][m][0], g1 = Gp[ic][m][1], g2 = Gp[ic][m][2];
        float g3 = Gp[ic][m][3], g4 = Gp[ic][m][4], g5 = Gp[ic][m][5];
        float g6 = Gp[ic][m][6];
        float* dst = hatG + (((bg * 32 + nc) * 1024 + p) << 3);
#pragma unroll
        for (int f = 0; f < 6; ++f) {
            dst[f] = g0 * W1t[f]      + g1 * W1t[6 + f]  + g2 * W1t[12 + f]
                   + g3 * W1t[18 + f] + g4 * W1t[24 + f] + g5 * W1t[30 + f]
                   + g6 * W1t[36 + f];
        }
    }
}

// ---------------------------------------------------------------------------
// K1: fully fused 3-iteration routing. One block per batch element b.
// 512 threads = 16 waves. Wave wv owns capsules nc = wv and wv+16; lane
// owns p = 32k+lane, k=0..31. t[nc][p] is exclusively owned by one thread,
// so it needs no init (iter 0 uses t==0 analytically) and no fences.
// Denominator D[q] (q = p & 127) reduced deterministically via LDS stages.
// ---------------------------------------------------------------------------
__global__ __launch_bounds__(512) void can_route_fused(
    const float* __restrict__ hatA, const float* __restrict__ hatG,
    float* __restrict__ tLog, float* __restrict__ out)
{
    __shared__ float Dpart[16][128];
    __shared__ float Dls[128];
    __shared__ float smOut[32][23];

    const int tid  = threadIdx.x;
    const int b    = blockIdx.x;
    const int lane = tid & 31;
    const int wv   = tid >> 5;

    float* tb = tLog + b * 32 * 1024;

    for (int iter = 0; iter < 3; ++iter) {
        // ---- denominator partials: d4[j] covers q = lane + 32j ----
        float d4[4] = {0.f, 0.f, 0.f, 0.f};
#pragma unroll 4
        for (int k = 0; k < 32; ++k) {
            float tv0 = 0.f, tv1 = 0.f;
            if (iter) {
                tv0 = tb[wv * 1024 + 32 * k + lane];
                tv1 = tb[(wv + 16) * 1024 + 32 * k + lane];
            }
            d4[k & 3] += expf(tv0) + expf(tv1);
        }
#pragma unroll
        for (int j = 0; j < 4; ++j) Dpart[wv][lane + 32 * j] = d4[j];
        __syncthreads();                     // Dpart ready; old Dls consumed
        if (tid < 128) {
            float s = 0.f;
#pragma unroll
            for (int w = 0; w < 16; ++w) s += Dpart[w][tid];
            Dls[tid] = 8.0f * s;             // NI=8 broadcast factor
        }
        __syncthreads();                     // Dls ready

        // ---- per-capsule accumulate + update ----
        for (int cap = 0; cap < 2; ++cap) {
            const int nc = wv + 16 * cap;
            const float* hA = hatA + ((size_t)(b * 32 + nc) << 14);
            const float* hG = hatG + ((size_t)(b * 32 + nc) << 13);
            float* tc = tb + nc * 1024;

            float accC = 0.f, accG[6], accA[16];
#pragma unroll
            for (int j = 0; j < 6; ++j)  accG[j] = 0.f;
#pragma unroll
            for (int j = 0; j < 16; ++j) accA[j] = 0.f;

#pragma unroll 4
            for (int k = 0; k < 32; ++k) {
                int p = 32 * k + lane;
                float tv = iter ? tc[p] : 0.f;
                float e = expf(tv) / Dls[lane + 32 * (k & 3)];
                accC += e;
                const float* gr = hG + (p << 3);
#pragma unroll
                for (int j = 0; j < 6; ++j)  accG[j] += e * gr[j];
                const float* ar = hA + (p << 4);
#pragma unroll
                for (int j = 0; j < 16; ++j) accA[j] += e * ar[j];
            }

            // wave32 butterfly reduce -> every lane holds the full sums
#pragma unroll
            for (int off = 16; off > 0; off >>= 1) {
                accC += __shfl_xor(accC, off, 32);
#pragma unroll
                for (int j = 0; j < 6; ++j)  accG[j] += __shfl_xor(accG[j], off, 32);
#pragma unroll
                for (int j = 0; j < 16; ++j) accA[j] += __shfl_xor(accA[j], off, 32);
            }

            float s = 0.f;
#pragma unroll
            for (int j = 0; j < 16; ++j) s += accA[j] * accA[j];
            float scale = s / (1.0f + s) / sqrtf(s + 1e-7f);
            float og[6];
#pragma unroll
            for (int j = 0; j < 6; ++j) og[j] = accG[j] / accC;   // mean_geom

            if (iter < 2) {
                // agree update: t += <og, hat_g> + 0.01 * <wattr, hat_a>
                for (int k = 0; k < 32; ++k) {
                    int p = 32 * k + lane;
                    const float* gr = hG + (p << 3);
                    const float* ar = hA + (p << 4);
                    float a1 = 0.f, a2 = 0.f;
#pragma unroll
                    for (int j = 0; j < 6; ++j)  a1 += og[j] * gr[j];
#pragma unroll
                    for (int j = 0; j < 16; ++j) a2 += accA[j] * ar[j];
                    float told = iter ? tc[p] : 0.f;
                    tc[p] = told + a1 + 0.01f * a2;
                }
            } else if (lane == 0) {
                smOut[nc][0] = scale;
#pragma unroll
                for (int j = 0; j < 6; ++j)  smOut[nc][1 + j] = og[j];
#pragma unroll
                for (int j = 0; j < 16; ++j) smOut[nc][7 + j] = accA[j];
            }
        }
    }

    __syncthreads();
    // output (B, NC, NI, DC); identical across NI -> replicate 8x
    for (int idx = tid; idx < 32 * 184; idx += 512) {
        int nc  = idx / 184;
        int rem = idx - nc * 184;
        int ni  = rem / 23;
        int d   = rem - ni * 23;
        out[((b * 32 + nc) * 8 + ni) * 23 + d] = smOut[nc][d];
    }
}

// ---------------------------------------------------------------------------
extern "C" void kernel_launch(void* const* d_in, const int* in_sizes, int n_in,
                              void* d_out, int out_size, void* d_ws, size_t ws_size,
                              hipStream_t stream)
{
    (void)in_sizes; (void)n_in; (void)out_size; (void)ws_size;
    const float* x  = (const float*)d_in[0];
    const float* W1 = (const float*)d_in[1];
    const float* W2 = (const float*)d_in[2];
    float* out = (float*)d_out;

    // Workspace partition (floats): hatA | hatG | t   (~52 MB, L2-resident)
    float* hatA = (float*)d_ws;                          // 16*32*1024*16
    float* hatG = hatA + (size_t)16 * 32 * 1024 * 16;    // 16*32*1024*8
    float* tLog = hatG + (size_t)16 * 32 * 1024 * 8;     // 16*32*1024

    can_hat_kernel<<<256, 128, 0, stream>>>(x, W1, W2, hatA, hatG);
    can_route_fused<<<16, 512, 0, stream>>>(hatA, hatG, tLog, out);
}